// DeltaNetDecodeWrapper_37417755083573
// MI455X (gfx1250) — compile-verified
//
#include <hip/hip_runtime.h>
#include <hip/hip_bf16.h>
#include <math.h>

// Problem constants (match reference)
#define H_   4096
#define HK_  16
#define DK_  256
#define VE_  512
#define QK_  4096   // HK*DK
#define VD_  8192   // HK*VE (= Hv*Dv)
#define CS_  4
#define EPS_ 1e-6f

typedef __attribute__((ext_vector_type(2))) float v2f;
typedef __attribute__((ext_vector_type(8))) float v8f;

__device__ __forceinline__ float sigmoid_(float x) { return 1.0f / (1.0f + __expf(-x)); }
__device__ __forceinline__ float silu_(float x)    { return x / (1.0f + __expf(-x)); }

// ---------------------------------------------------------------------------
// Kernel 1: all input-projection matvecs. One 256-thread block per weight row.
// Rows: [0,4096) Wq -> q, [4096,8192) Wk -> k, [8192,16384) Wv -> v,
//       [16384,16400) Wa -> ab[0..16), [16400,16416) Wb -> ab[16..32)
// Each row is 4096 fp32 = 1024 float4; thread t handles float4 idx {t, t+256, t+512, t+768}
// -> fully coalesced 4 KB per warp-pass; h stays hot in L2 (192 MB).
// ---------------------------------------------------------------------------
__global__ __launch_bounds__(256)
void k_qkvab(const float* __restrict__ h,
             const float* __restrict__ Wq, const float* __restrict__ Wk,
             const float* __restrict__ Wv, const float* __restrict__ Wa,
             const float* __restrict__ Wb,
             float* __restrict__ q, float* __restrict__ k,
             float* __restrict__ v, float* __restrict__ ab)
{
    int b = blockIdx.x;
    const float* W; float* out; int row;
    if      (b < 4096)  { W = Wq; out = q;      row = b;         }
    else if (b < 8192)  { W = Wk; out = k;      row = b - 4096;  }
    else if (b < 16384) { W = Wv; out = v;      row = b - 8192;  }
    else if (b < 16400) { W = Wa; out = ab;     row = b - 16384; }
    else                { W = Wb; out = ab + HK_; row = b - 16400; }

    const float4* wrow = (const float4*)(W + (size_t)row * H_);
    const float4* hv   = (const float4*)h;

    int t = threadIdx.x;
    float acc = 0.0f;
#pragma unroll
    for (int p = 0; p < 4; ++p) {
        int idx = p * 256 + t;
        float4 w4 = wrow[idx];
        float4 h4 = hv[idx];
        acc += w4.x * h4.x + w4.y * h4.y + w4.z * h4.z + w4.w * h4.w;
    }

    __shared__ float sdata[256];
    sdata[t] = acc;
    __syncthreads();
    for (int s = 128; s > 0; s >>= 1) {
        if (t < s) sdata[t] += sdata[t + s];
        __syncthreads();
    }
    if (t == 0) out[row] = sdata[0];
}

// ---------------------------------------------------------------------------
// Kernel 2: depthwise conv (CS=4) + SiLU, per-head L2-norm for q/k, sigmoid a/b.
// Blocks: [0,16) q-heads, [16,32) k-heads, [32,64) v chunks of 256, 64: alpha/beta.
// ---------------------------------------------------------------------------
__global__ __launch_bounds__(256)
void k_conv(const float* __restrict__ q,  const float* __restrict__ kk,
            const float* __restrict__ v,  const float* __restrict__ ab,
            const float* __restrict__ qw, const float* __restrict__ kw,
            const float* __restrict__ vw,
            const float* __restrict__ qc, const float* __restrict__ kc,
            const float* __restrict__ vc,
            float* __restrict__ q1n, float* __restrict__ k1n,
            float* __restrict__ v1,  float* __restrict__ alp,
            float* __restrict__ bet)
{
    int b = blockIdx.x;
    int t = threadIdx.x;
    __shared__ float sdata[256];

    if (b == 64) {
        if (t < HK_)            alp[t]        = sigmoid_(ab[t]);
        else if (t < 2 * HK_)   bet[t - HK_]  = sigmoid_(ab[t]);
        return;
    }

    if (b < 32) {
        // q (b<16) or k (16<=b<32) head: conv+silu then l2norm over 256 dims
        bool isQ = (b < HK_);
        int head = isQ ? b : (b - HK_);
        int i = head * DK_ + t;
        const float* x0 = isQ ? q  : kk;
        const float* cw = isQ ? qw : kw;
        const float* cc = isQ ? qc : kc;
        float x = cc[i * 3 + 0] * cw[i * 4 + 0]
                + cc[i * 3 + 1] * cw[i * 4 + 1]
                + cc[i * 3 + 2] * cw[i * 4 + 2]
                + x0[i]         * cw[i * 4 + 3];
        x = silu_(x);
        sdata[t] = x * x;
        __syncthreads();
        for (int s = 128; s > 0; s >>= 1) {
            if (t < s) sdata[t] += sdata[t + s];
            __syncthreads();
        }
        float inv = rsqrtf(sdata[0] + EPS_);
        (isQ ? q1n : k1n)[i] = x * inv;
    } else {
        // v chunk: conv + silu only
        int i = (b - 32) * 256 + t;
        float x = vc[i * 3 + 0] * vw[i * 4 + 0]
                + vc[i * 3 + 1] * vw[i * 4 + 1]
                + vc[i * 3 + 2] * vw[i * 4 + 2]
                + v[i]          * vw[i * 4 + 3];
        v1[i] = silu_(x);
    }
}

// ---------------------------------------------------------------------------
// Kernel 3 (WMMA): per head, single pass over S (256x512) computes BOTH
//   s_q = q^T S   and   s_k = k^T S
// via v_wmma_f32_16x16x4_f32 with A = [q;k;0...] (16x4 K-slices), B = S tile
// (4x16). Then out[v] = a*s_q[v] + b*(q.k)*(v1[v] - a*s_k[v])  -- new_state is
// never materialized (saves the 8.4 MB state write entirely).
//
// A layout (ISA 7.12.2, f32 16x4): M = lane&15, K = vgpr + 2*(lane>>4).
// B layout (4x16, row striped across lanes): N = lane&15, K = vgpr + 2*(lane>>4).
// C/D: row j in VGPR j for lanes 0-15.
// One wave (32 threads) per (head, 16-column tile); EXEC all-ones around WMMA.
// ---------------------------------------------------------------------------
__global__ __launch_bounds__(32)
void k_state(const float* __restrict__ state,
             const float* __restrict__ q1n, const float* __restrict__ k1n,
             const float* __restrict__ v1,
             const float* __restrict__ alp, const float* __restrict__ bet,
             float* __restrict__ ov)
{
    const int head = blockIdx.y;
    const int n0   = blockIdx.x * 16;
    const int lane = threadIdx.x;
    const int half = lane >> 4;     // 0: lanes 0-15, 1: lanes 16-31
    const int m    = lane & 15;     // A row / B column / D column

    __shared__ float lq[DK_];
    __shared__ float lk[DK_];
#pragma unroll
    for (int j = 0; j < DK_ / 32; ++j) {
        lq[lane + 32 * j] = q1n[head * DK_ + lane + 32 * j];
        lk[lane + 32 * j] = k1n[head * DK_ + lane + 32 * j];
    }
    __syncthreads();

    // q.k dot (all lanes end with the value)
    float qkp = 0.0f;
#pragma unroll
    for (int j = 0; j < DK_ / 32; ++j) qkp += lq[lane + 32 * j] * lk[lane + 32 * j];
#pragma unroll
    for (int off = 16; off > 0; off >>= 1) qkp += __shfl_xor(qkp, off, 32);
    const float qk = qkp;

    // A-row source: row 0 = q, row 1 = k, rows 2..15 = 0 (branchless select)
    const float  asel = (m < 2) ? 1.0f : 0.0f;
    const float* arow = (m == 1) ? lk : lq;

    const float* Sh   = state + (size_t)head * DK_ * VE_;
    const float* bptr = Sh + (size_t)(2 * half) * VE_ + n0 + m;  // row 2*half, col n0+m

    v8f acc = {};
#pragma unroll 4
    for (int d0 = 0; d0 < DK_; d0 += 4) {
        int ai = d0 + 2 * half;
        v2f a, b;
        a.x = asel * arow[ai];
        a.y = asel * arow[ai + 1];
        b.x = bptr[(size_t)d0 * VE_];          // K = d0 + 2*half
        b.y = bptr[(size_t)d0 * VE_ + VE_];    // K = d0 + 2*half + 1
        acc = __builtin_amdgcn_wmma_f32_16x16x4_f32(
                  false, a, false, b, (short)0, acc, false, false);
    }

    // D row 0 = s_q chunk, row 1 = s_k chunk (lanes 0-15, N = lane)
    if (lane < 16) {
        float a_s = alp[head];
        float b_s = bet[head];
        float vh  = v1[head * VE_ + n0 + lane];
        float s_q = acc[0];
        float s_k = acc[1];
        ov[head * VE_ + n0 + lane] = a_s * s_q + b_s * qk * (vh - a_s * s_k);
    }
}

// ---------------------------------------------------------------------------
// Kernel 4: output projection. One block per output row of Wo (4096 x 8192).
// Row = 8192 fp32 = 2048 float4; thread t handles 8 float4s, coalesced.
// ---------------------------------------------------------------------------
__global__ __launch_bounds__(256)
void k_outproj(const float* __restrict__ Wo, const float* __restrict__ ov,
               float* __restrict__ out)
{
    int row = blockIdx.x;
    int t   = threadIdx.x;
    const float4* wrow = (const float4*)(Wo + (size_t)row * VD_);
    const float4* xv   = (const float4*)ov;
    float acc = 0.0f;
#pragma unroll
    for (int p = 0; p < 8; ++p) {
        int idx = p * 256 + t;
        float4 w4 = wrow[idx];
        float4 x4 = xv[idx];
        acc += w4.x * x4.x + w4.y * x4.y + w4.z * x4.z + w4.w * x4.w;
    }
    __shared__ float sdata[256];
    sdata[t] = acc;
    __syncthreads();
    for (int s = 128; s > 0; s >>= 1) {
        if (t < s) sdata[t] += sdata[t + s];
        __syncthreads();
    }
    if (t == 0) out[row] = sdata[0];
}

// ---------------------------------------------------------------------------
extern "C" void kernel_launch(void* const* d_in, const int* in_sizes, int n_in,
                              void* d_out, int out_size, void* d_ws, size_t ws_size,
                              hipStream_t stream)
{
    const float* h   = (const float*)d_in[0];   // (1,H,1,1) -> 4096
    const float* Wq  = (const float*)d_in[1];   // (QK,H)
    const float* Wk  = (const float*)d_in[2];   // (QK,H)
    const float* Wv  = (const float*)d_in[3];   // (VD,H)
    const float* Wo  = (const float*)d_in[4];   // (H,VD)
    const float* Wa  = (const float*)d_in[5];   // (Hk,H)
    const float* Wb  = (const float*)d_in[6];   // (Hk,H)
    const float* qw  = (const float*)d_in[7];   // (1,QK,CS)
    const float* kw  = (const float*)d_in[8];   // (1,QK,CS)
    const float* vw  = (const float*)d_in[9];   // (1,VD,CS)
    const float* qc  = (const float*)d_in[10];  // (1,QK,CS-1)
    const float* kc  = (const float*)d_in[11];  // (1,QK,CS-1)
    const float* vc  = (const float*)d_in[12];  // (1,VD,CS-1)
    const float* st  = (const float*)d_in[13];  // (1,Hk,Dk,Ve)
    float* out = (float*)d_out;                 // (1,H,1,1) fp32

    float* ws  = (float*)d_ws;
    float* q   = ws;            // 4096
    float* k   = q   + QK_;     // 4096
    float* v   = k   + QK_;     // 8192
    float* ab  = v   + VD_;     // 32 (alpha_lin | beta_lin)
    float* q1n = ab  + 32;      // 4096
    float* k1n = q1n + QK_;     // 4096
    float* v1  = k1n + QK_;     // 8192
    float* alp = v1  + VD_;     // 16
    float* bet = alp + HK_;     // 16
    float* ov  = bet + HK_;     // 8192  (fused out vector, reshaped (1,VD))

    // 1) input projections: 4096+4096+8192+16+16 = 16416 rows
    k_qkvab<<<dim3(16416), dim3(256), 0, stream>>>(h, Wq, Wk, Wv, Wa, Wb, q, k, v, ab);

    // 2) conv + silu + l2norm + sigmoid
    k_conv<<<dim3(65), dim3(256), 0, stream>>>(q, k, v, ab, qw, kw, vw, qc, kc, vc,
                                               q1n, k1n, v1, alp, bet);

    // 3) WMMA state pass: grid (Ve/16 column tiles, Hk heads), 1 wave each
    k_state<<<dim3(VE_ / 16, HK_), dim3(32), 0, stream>>>(st, q1n, k1n, v1, alp, bet, ov);

    // 4) output projection
    k_outproj<<<dim3(H_), dim3(256), 0, stream>>>(Wo, ov, out);
}